// LIFLayer_78640851190231
// MI455X (gfx1250) — compile-verified
//
#include <hip/hip_runtime.h>

// LIF recurrence: V = decay*V + I; spike = V >= v_th; V = spike ? v_reset : V
// input (T=256, B=32, N=4096) fp32 -> outputs [spike_trace | voltage_trace], each T*B*N fp32.
//
// HBM-bound streaming kernel (~402 MB total traffic, AI ~0.33 flop/byte -> ~17us roofline).
//  - one thread owns 4 contiguous neurons; V lives in VGPRs across all 256 steps
//  - gfx1250 async copy engine (global_load_async_to_lds_b128 + s_wait_asynccnt)
//    double-buffers the next timestep's tile; T-loop unrolled x2 so buffer
//    alternation is static (no per-iteration LDS address math)
//  - each lane consumes exactly the 16B it staged -> ASYNCcnt wait only, no barrier
//  - 128-bit nontemporal stores for the two write-once output streams

typedef float v4f __attribute__((ext_vector_type(4)));

#define LIF_T    256
#define LIF_B    32
#define LIF_N    4096
#define LIF_BN   (LIF_B * LIF_N)        // 131072 neurons
#define THREADS  128                    // 4 wave32 per workgroup
#define PER_THR  4                      // neurons per thread (float4)
#define COLS     (THREADS * PER_THR)    // 512 neurons per block
#define BLOCKS   (LIF_BN / COLS)        // 256 workgroups -> 1024 waves in flight

__global__ __launch_bounds__(THREADS) void lif_stream_kernel(
    const float* __restrict__ in,       // (T, B*N)
    const float* __restrict__ p_decay,
    const float* __restrict__ p_vth,
    const float* __restrict__ p_vreset,
    float* __restrict__ out_spike,      // (T, B*N)
    float* __restrict__ out_volt)       // (T, B*N)
{
    __shared__ v4f buf[2][THREADS];     // 2 x 2KB double buffer

    const float decay = p_decay[0];
    const float v_th  = p_vth[0];
    const float v_rst = p_vreset[0];

    const int tid  = threadIdx.x;
    const int col0 = blockIdx.x * COLS + tid * PER_THR;   // first of 4 neurons

    // Wave-relative LDS byte offsets for this lane's slot in each buffer
    // (low 32 bits of the flat shared-aperture address == LDS offset).
    const unsigned lds0 = (unsigned)(size_t)&buf[0][tid];
    const unsigned lds1 = (unsigned)(size_t)&buf[1][tid];

    const unsigned gstride = (unsigned)(LIF_BN * sizeof(float)); // bytes per timestep

    v4f V = {0.0f, 0.0f, 0.0f, 0.0f};
    float* ps = out_spike + col0;
    float* pv = out_volt  + col0;

    auto issue = [&](unsigned lds, unsigned go) {
        asm volatile("global_load_async_to_lds_b128 %0, %1, %2"
                     :: "v"(lds), "v"(go), "s"(in) : "memory");
    };
    auto step = [&](const v4f I) {
        v4f Vn = decay * V + I;                 // leak + integrate (packed FMA)
        v4f spk;
        spk.x = (Vn.x >= v_th) ? 1.0f : 0.0f;
        spk.y = (Vn.y >= v_th) ? 1.0f : 0.0f;
        spk.z = (Vn.z >= v_th) ? 1.0f : 0.0f;
        spk.w = (Vn.w >= v_th) ? 1.0f : 0.0f;
        V.x = (Vn.x >= v_th) ? v_rst : Vn.x;    // post-reset value == trace value
        V.y = (Vn.y >= v_th) ? v_rst : Vn.y;
        V.z = (Vn.z >= v_th) ? v_rst : Vn.z;
        V.w = (Vn.w >= v_th) ? v_rst : Vn.w;
        __builtin_nontemporal_store(spk, (v4f*)ps);
        __builtin_nontemporal_store(V,   (v4f*)pv);
        ps += LIF_BN;
        pv += LIF_BN;
    };

    // Prologue: stage timestep 0 into buffer 0.
    unsigned go = (unsigned)(col0 * sizeof(float));
    issue(lds0, go);
    go += gstride;                              // offset of timestep 1

    // Steady state: t = 0..253, unrolled x2, static buffer alternation.
    // Async loads complete in order, so asynccnt<=1 => the older copy landed.
    for (int it = 0; it < (LIF_T - 2) / 2; ++it) {
        issue(lds1, go);                        // L(t+1) -> buf1
        asm volatile("s_wait_asynccnt 0x1" ::: "memory");
        step(buf[0][tid]);                      // consume t (even)

        issue(lds0, go + gstride);              // L(t+2) -> buf0
        asm volatile("s_wait_asynccnt 0x1" ::: "memory");
        step(buf[1][tid]);                      // consume t+1 (odd)

        go += 2 * gstride;
    }

    // Epilogue: t = 254, 255.
    issue(lds1, go);                            // L255 -> buf1
    asm volatile("s_wait_asynccnt 0x1" ::: "memory");
    step(buf[0][tid]);                          // t = 254
    asm volatile("s_wait_asynccnt 0x0" ::: "memory");
    step(buf[1][tid]);                          // t = 255
}

extern "C" void kernel_launch(void* const* d_in, const int* in_sizes, int n_in,
                              void* d_out, int out_size, void* d_ws, size_t ws_size,
                              hipStream_t stream) {
    const float* in       = (const float*)d_in[0];
    const float* p_decay  = (const float*)d_in[1];
    const float* p_vth    = (const float*)d_in[2];
    const float* p_vreset = (const float*)d_in[3];

    float* out_spike = (float*)d_out;
    float* out_volt  = out_spike + (size_t)LIF_T * LIF_BN;

    lif_stream_kernel<<<BLOCKS, THREADS, 0, stream>>>(
        in, p_decay, p_vth, p_vreset, out_spike, out_volt);
}